// DeformationOptModule_80204219285780
// MI455X (gfx1250) — compile-verified
//
#include <hip/hip_runtime.h>
#include <hip/hip_bf16.h>
#include <math.h>

typedef __attribute__((ext_vector_type(2))) float v2f;
typedef __attribute__((ext_vector_type(8))) float v8f;

#define NPB   64      // points per block
#define NTHR  128     // 4 wave32s
#define HSTR  68      // LDS feature-row stride (floats), bank-conflict padded
#define WSTR  66      // LDS weight-row stride (floats), bank-conflict padded

struct KParams {
    const float* means;
    const float* quats;
    const float* itime;
    const float* grids[18];
    const float* gT;                 // channel-last transposed grids (in d_ws), or null
    unsigned long long gofs[18];     // float offsets into gT
    const float* W[5];
    const float* Bv[5];
    float* out0;                     // means + delta   [N*3]
    float* out1;                     // quats_n + delta [N*4]
    int N;
};

__device__ __forceinline__ void load8(float* d, const float* p) {
    const float4 a = *reinterpret_cast<const float4*>(p);
    const float4 b = *reinterpret_cast<const float4*>(p + 4);
    d[0] = a.x; d[1] = a.y; d[2] = a.z; d[3] = a.w;
    d[4] = b.x; d[5] = b.y; d[6] = b.z; d[7] = b.w;
}

// Transpose one grid [16,H,W] -> [H,W,16] (channel-last) into workspace.
__global__ void transpose_grid_kernel(const float* __restrict__ in,
                                      float* __restrict__ out, int HW) {
    int t = blockIdx.x * 256 + threadIdx.x;
    if (t < HW * 16) {
        int c  = t & 15;
        int hw = t >> 4;
        out[t] = in[(size_t)c * HW + hw];
    }
}

// Cooperative weight staging: row-major [din][DOUT] -> LDS with row stride WSTR.
template <int DOUT>
__device__ __forceinline__ void stage_weights(float* Wst, const float* Wg,
                                              int din, int tid) {
    const int nel = din * DOUT;
    for (int idx = tid; idx < nel; idx += NTHR) {
        int k = idx / DOUT;          // compile-time divisor
        int n = idx - k * DOUT;
        Wst[k * WSTR + n] = Wg[idx];
    }
}

// Fully unrolled 16xDOUT tile GEMM on V_WMMA_F32_16X16X4_F32.
// A (16 x 4*KSTEPS) from hin (LDS, stride HSTR); B from Wst (LDS, stride WSTR).
template <int KSTEPS, int NTILES>
__device__ __forceinline__ void wmma_tiles(const float* hin, const float* Wst,
                                           int lm, int half, v8f acc[NTILES]) {
    const v8f vzero = {0.f, 0.f, 0.f, 0.f, 0.f, 0.f, 0.f, 0.f};
    #pragma unroll
    for (int nt = 0; nt < NTILES; ++nt) acc[nt] = vzero;
    #pragma unroll
    for (int kk = 0; kk < KSTEPS; ++kk) {
        const int kb = kk * 4 + half * 2;   // K index of VGPR j is kb+j
        v2f a;
        a[0] = hin[lm * HSTR + kb];
        a[1] = hin[lm * HSTR + kb + 1];
        #pragma unroll
        for (int nt = 0; nt < NTILES; ++nt) {
            v2f b;
            b[0] = Wst[kb * WSTR + nt * 16 + lm];
            b[1] = Wst[(kb + 1) * WSTR + nt * 16 + lm];
            acc[nt] = __builtin_amdgcn_wmma_f32_16x16x4_f32(
                false, a, false, b, (short)0, acc[nt], false, false);
        }
    }
}

// Bias + ReLU epilogue: D(VGPR r, lane) -> hout[M = r+8*half][N = nt*16+lm].
template <int NTILES>
__device__ __forceinline__ void epilogue_relu(const v8f acc[NTILES],
                                              const float* bias_g, float* hout,
                                              int lm, int half) {
    #pragma unroll
    for (int nt = 0; nt < NTILES; ++nt) {
        const float bias = bias_g[nt * 16 + lm];
        #pragma unroll
        for (int r = 0; r < 8; ++r) {
            float v = fmaxf(acc[nt][r] + bias, 0.0f);
            hout[(r + 8 * half) * HSTR + nt * 16 + lm] = v;
        }
    }
}

template <int CHLAST>
__global__ __launch_bounds__(NTHR) void deform_kernel(KParams P) {
    __shared__ float hA[4][16][HSTR];
    __shared__ float hB[4][16][HSTR];
    __shared__ float Wst[64 * WSTR];

    const int tid  = threadIdx.x;
    const int lane = tid & 31;
    const int wv   = tid >> 5;
    const int base = blockIdx.x * NPB;

    // ---------------- Phase A: hexplane features -> LDS ----------------
    {
        const int ptl = tid >> 1;        // local point 0..63 (wave-private tile)
        const int m   = ptl & 15;
        const int chh = tid & 1;         // channel half
        const int c0  = chh * 8;
        const int gpt = base + ptl;

        if (gpt < P.N) {
            float pc[4];
            #pragma unroll
            for (int a = 0; a < 3; ++a) {
                float x = P.means[(size_t)gpt * 3 + a];
                float v = copysignf(log1pf(fabsf(x)), x) / 5.0f;
                pc[a] = fminf(fmaxf(v, -1.0f), 1.0f);
            }
            pc[3] = P.itime[0];

            int gp = 0;
            #pragma unroll
            for (int L = 0; L < 3; ++L) {
                const int SW = 64 << L;
                float prod[8];
                #pragma unroll
                for (int c = 0; c < 8; ++c) prod[c] = 1.0f;

                #pragma unroll
                for (int q = 0; q < 6; ++q, ++gp) {
                    constexpr int QI[6] = {0, 0, 0, 1, 1, 2};
                    constexpr int QJ[6] = {1, 2, 3, 2, 3, 3};
                    const int Wd = SW;
                    const int Hd = (QJ[q] == 3) ? 150 : SW;
                    const float cx = pc[QI[q]];
                    const float cy = pc[QJ[q]];
                    float xf  = (cx + 1.0f) * 0.5f * (float)(Wd - 1);
                    float yf  = (cy + 1.0f) * 0.5f * (float)(Hd - 1);
                    float x0f = fminf(fmaxf(floorf(xf), 0.0f), (float)(Wd - 1));
                    float y0f = fminf(fmaxf(floorf(yf), 0.0f), (float)(Hd - 1));
                    const float wx = xf - x0f, wy = yf - y0f;
                    const int x0 = (int)x0f, y0 = (int)y0f;
                    const int x1 = min(x0 + 1, Wd - 1);
                    const int y1 = min(y0 + 1, Hd - 1);
                    const float w00 = (1.0f - wx) * (1.0f - wy);
                    const float w01 = wx * (1.0f - wy);
                    const float w10 = (1.0f - wx) * wy;
                    const float w11 = wx * wy;

                    if (CHLAST) {
                        const float* g = P.gT + P.gofs[gp];
                        float v00[8], v01[8], v10[8], v11[8];
                        load8(v00, g + ((size_t)((size_t)y0 * Wd + x0) * 16 + c0));
                        load8(v01, g + ((size_t)((size_t)y0 * Wd + x1) * 16 + c0));
                        load8(v10, g + ((size_t)((size_t)y1 * Wd + x0) * 16 + c0));
                        load8(v11, g + ((size_t)((size_t)y1 * Wd + x1) * 16 + c0));
                        #pragma unroll
                        for (int c = 0; c < 8; ++c)
                            prod[c] *= v00[c] * w00 + v01[c] * w01 +
                                       v10[c] * w10 + v11[c] * w11;
                    } else {
                        const float* g = P.grids[gp];
                        const size_t HW = (size_t)Hd * Wd;
                        #pragma unroll
                        for (int c = 0; c < 8; ++c) {
                            const float* gc = g + (size_t)(c0 + c) * HW;
                            float v00 = gc[(size_t)y0 * Wd + x0];
                            float v01 = gc[(size_t)y0 * Wd + x1];
                            float v10 = gc[(size_t)y1 * Wd + x0];
                            float v11 = gc[(size_t)y1 * Wd + x1];
                            prod[c] *= v00 * w00 + v01 * w01 + v10 * w10 + v11 * w11;
                        }
                    }
                }
                #pragma unroll
                for (int c = 0; c < 8; ++c)
                    hA[wv][m][L * 16 + c0 + c] = prod[c];
            }
        } else {
            #pragma unroll
            for (int L = 0; L < 3; ++L)
                #pragma unroll
                for (int c = 0; c < 8; ++c)
                    hA[wv][m][L * 16 + c0 + c] = 0.0f;
        }
    }

    // ---------------- Phase B: MLP via V_WMMA_F32_16X16X4_F32 ----------------
    const int half = lane >> 4;          // 0: lanes 0-15, 1: lanes 16-31
    const int lm   = lane & 15;
    float* hin  = &hA[wv][0][0];
    float* hout = &hB[wv][0][0];

    // ---- layer 0: 48 -> 64, ReLU ----
    stage_weights<64>(Wst, P.W[0], 48, tid);
    __syncthreads();
    {
        v8f acc[4];
        wmma_tiles<12, 4>(hin, Wst, lm, half, acc);
        epilogue_relu<4>(acc, P.Bv[0], hout, lm, half);
        float* t = hin; hin = hout; hout = t;
    }

    // ---- layers 1..3: 64 -> 64, ReLU ----
    for (int l = 1; l < 4; ++l) {
        __syncthreads();
        stage_weights<64>(Wst, P.W[l], 64, tid);
        __syncthreads();
        v8f acc[4];
        wmma_tiles<16, 4>(hin, Wst, lm, half, acc);
        epilogue_relu<4>(acc, P.Bv[l], hout, lm, half);
        float* t = hin; hin = hout; hout = t;
    }

    // ---- layer 4: 64 -> 7, linear, fused output epilogue ----
    __syncthreads();
    stage_weights<7>(Wst, P.W[4], 64, tid);
    __syncthreads();
    {
        v8f acc[1];
        wmma_tiles<16, 1>(hin, Wst, lm, half, acc);

        const int n = lm;
        const float bias = (n < 7) ? P.Bv[4][n] : 0.0f;
        #pragma unroll
        for (int r = 0; r < 8; ++r) {
            const float v  = acc[0][r] + bias;
            const int   m  = r + 8 * half;
            const int   Pg = base + wv * 16 + m;
            if (Pg < P.N) {
                if (n < 3) {
                    P.out0[(size_t)Pg * 3 + n] = P.means[(size_t)Pg * 3 + n] + v;
                } else if (n < 7) {
                    const float q0 = P.quats[(size_t)Pg * 4 + 0];
                    const float q1 = P.quats[(size_t)Pg * 4 + 1];
                    const float q2 = P.quats[(size_t)Pg * 4 + 2];
                    const float q3 = P.quats[(size_t)Pg * 4 + 3];
                    const float nrm = sqrtf(q0 * q0 + q1 * q1 + q2 * q2 + q3 * q3);
                    P.out1[(size_t)Pg * 4 + (n - 3)] =
                        P.quats[(size_t)Pg * 4 + (n - 3)] / nrm + v;
                }
            }
        }
    }
}

extern "C" void kernel_launch(void* const* d_in, const int* in_sizes, int n_in,
                              void* d_out, int out_size, void* d_ws, size_t ws_size,
                              hipStream_t stream) {
    // Input order: 0 means, 1 quats, 2 sh0, 3 features, 4 render_times,
    //              5 image_times, 6..23 grids, 24..33 (W0,b0,...,W4,b4)
    const int N = in_sizes[0] / 3;

    static const int QJ[6] = {1, 2, 3, 2, 3, 3};
    int Hd[18], Wd[18];
    unsigned long long ofs[18];
    unsigned long long tot = 0;
    for (int L = 0; L < 3; ++L) {
        const int SW = 64 << L;
        for (int q = 0; q < 6; ++q) {
            const int p = L * 6 + q;
            Wd[p] = SW;
            Hd[p] = (QJ[q] == 3) ? 150 : SW;
            ofs[p] = tot;
            tot += (unsigned long long)16 * Hd[p] * Wd[p];
        }
    }
    const bool chlast = (ws_size >= tot * sizeof(float));

    KParams P;
    P.means = (const float*)d_in[0];
    P.quats = (const float*)d_in[1];
    P.itime = (const float*)d_in[5];
    for (int p = 0; p < 18; ++p) {
        P.grids[p] = (const float*)d_in[6 + p];
        P.gofs[p]  = ofs[p];
    }
    P.gT = chlast ? (const float*)d_ws : nullptr;
    for (int l = 0; l < 5; ++l) {
        P.W[l]  = (const float*)d_in[24 + 2 * l];
        P.Bv[l] = (const float*)d_in[25 + 2 * l];
    }
    P.out0 = (float*)d_out;
    P.out1 = (float*)d_out + (size_t)3 * N;
    P.N    = N;

    if (chlast) {
        float* ws = (float*)d_ws;
        for (int p = 0; p < 18; ++p) {
            const int HW = Hd[p] * Wd[p];
            const int nel = HW * 16;
            transpose_grid_kernel<<<(nel + 255) / 256, 256, 0, stream>>>(
                (const float*)d_in[6 + p], ws + ofs[p], HW);
        }
        deform_kernel<1><<<(N + NPB - 1) / NPB, NTHR, 0, stream>>>(P);
    } else {
        deform_kernel<0><<<(N + NPB - 1) / NPB, NTHR, 0, stream>>>(P);
    }
}